// GCNModel_79285096284187
// MI455X (gfx1250) — compile-verified
//
#include <hip/hip_runtime.h>
#include <hip/hip_bf16.h>

#define HID 128
#define BN_EPS 1e-5f

typedef float v2f __attribute__((ext_vector_type(2)));
typedef float v8f __attribute__((ext_vector_type(8)));

// ======================= graph preprocessing =======================

// In-degree histogram over dst (edge_index row 1).
__global__ void k_hist(const int* __restrict__ ei, int E, int* __restrict__ counts) {
  int e = blockIdx.x * blockDim.x + threadIdx.x;
  if (e < E) atomicAdd(&counts[ei[E + e]], 1);
}

// deg = indeg + 1 (self loop); dinv = 1/sqrt(deg); selfn = 1/deg.
__global__ void k_deg(const int* __restrict__ counts, int N,
                      float* __restrict__ dinv, float* __restrict__ selfn) {
  int v = blockIdx.x * blockDim.x + threadIdx.x;
  if (v < N) {
    float d = (float)(counts[v] + 1);
    dinv[v]  = rsqrtf(d);
    selfn[v] = 1.0f / d;
  }
}

// Exclusive prefix sum of counts -> offsets (single workgroup, chunked Hillis-Steele).
__global__ void k_scan(const int* __restrict__ counts, int* __restrict__ offsets, int n) {
  __shared__ int sh[1024];
  int tid = threadIdx.x;
  int carry = 0;
  for (int base = 0; base < n; base += 1024) {
    int idx = base + tid;
    int v = (idx < n) ? counts[idx] : 0;
    sh[tid] = v;
    __syncthreads();
    for (int off = 1; off < 1024; off <<= 1) {
      int t = (tid >= off) ? sh[tid - off] : 0;
      __syncthreads();
      sh[tid] += t;
      __syncthreads();
    }
    if (idx < n) offsets[idx] = carry + sh[tid] - v;
    carry += sh[1023];
    __syncthreads();
  }
  if (tid == 0) offsets[n] = carry;
}

// Fill CSR: per-edge slot = offsets[dst] + atomic cursor; precompute enorm per slot.
__global__ void k_fill(const int* __restrict__ ei, int E, const int* __restrict__ offsets,
                       int* __restrict__ cursor, const float* __restrict__ dinv,
                       int* __restrict__ csr_src, float* __restrict__ csr_w) {
  int e = blockIdx.x * blockDim.x + threadIdx.x;
  if (e >= E) return;
  int s = ei[e];
  int d = ei[E + e];
  int pos = atomicAdd(&cursor[d], 1);
  int slot = offsets[d] + pos;
  csr_src[slot] = s;
  csr_w[slot] = dinv[s] * dinv[d];
}

// Fold eval-mode BatchNorm into per-feature scale/shift for all 4 layers.
__global__ void k_bnprep(const float* __restrict__ g, const float* __restrict__ b,
                         const float* __restrict__ m, const float* __restrict__ v,
                         float* __restrict__ scale, float* __restrict__ shift) {
  int i = blockIdx.x * blockDim.x + threadIdx.x;
  if (i < 4 * HID) {
    float inv = rsqrtf(v[i] + BN_EPS);
    float sc = g[i] * inv;
    scale[i] = sc;
    shift[i] = b[i] - m[i] * sc;
  }
}

// ======================= fp32 WMMA GEMM: C[N,128] = A[N,128] @ W[128,128] ==========
// One wave per 16x16 C tile; 8 waves/block share a 16x128 A strip staged in LDS.
// f32 fragment layout (ISA 7.12.2): lanes 0-15 carry K=k..k+1, lanes 16-31 K=k+2..k+3.
__global__ __launch_bounds__(256) void k_gemm(const float* __restrict__ A,
                                              const float* __restrict__ W,
                                              float* __restrict__ C) {
  __shared__ float sA[16 * 132];  // pad stride 132 -> conflict-free column reads
  int tid = threadIdx.x;
  int row0 = blockIdx.x * 16;
  for (int i = tid; i < 16 * HID; i += 256) {
    int r = i >> 7, c = i & 127;
    sA[r * 132 + c] = A[(size_t)(row0 + r) * HID + c];
  }
  __syncthreads();

  int lane = tid & 31, wave = tid >> 5;
  int half = lane >> 4;
  int mn = lane & 15;          // A row / B,C column within tile
  int colbase = wave * 16;     // 8 waves cover all 128 output columns
  v8f acc = {0.f, 0.f, 0.f, 0.f, 0.f, 0.f, 0.f, 0.f};

#pragma unroll
  for (int k = 0; k < HID; k += 4) {
    int ka = k + 2 * half;
    v2f a, b;
    a.x = sA[mn * 132 + ka];
    a.y = sA[mn * 132 + ka + 1];
    b.x = W[(size_t)ka * HID + colbase + mn];
    b.y = W[(size_t)(ka + 1) * HID + colbase + mn];
    acc = __builtin_amdgcn_wmma_f32_16x16x4_f32(false, a, false, b,
                                                (short)0, acc, false, false);
  }

  int mbase = half * 8;  // C layout: lanes 0-15 rows 0..7, lanes 16-31 rows 8..15
#pragma unroll
  for (int r = 0; r < 8; ++r)
    C[(size_t)(row0 + mbase + r) * HID + colbase + mn] = acc[r];
}

// ======================= CSR aggregation + bias + relu + BN =======================
// One wave per node; each lane owns a float4 feature slice (32*4 = 128).
__global__ __launch_bounds__(256) void k_aggregate(
    const float* __restrict__ t, const int* __restrict__ offsets,
    const int* __restrict__ csr_src, const float* __restrict__ csr_w,
    const float* __restrict__ selfn, const float* __restrict__ bias,
    const float* __restrict__ scale, const float* __restrict__ shift,
    float* __restrict__ h, int N) {
  int wave = threadIdx.x >> 5, lane = threadIdx.x & 31;
  int v = blockIdx.x * 8 + wave;
  if (v >= N) return;

  float sn = selfn[v];
  float4 acc = ((const float4*)(t + (size_t)v * HID))[lane];
  acc.x *= sn; acc.y *= sn; acc.z *= sn; acc.w *= sn;

  int beg = offsets[v], end = offsets[v + 1];
  for (int j = beg; j < end; ++j) {
    int s = csr_src[j];
    float w = csr_w[j];
    float4 r = ((const float4*)(t + (size_t)s * HID))[lane];
    acc.x += r.x * w; acc.y += r.y * w; acc.z += r.z * w; acc.w += r.w * w;
  }

  float4 bi = ((const float4*)bias)[lane];
  float4 sc = ((const float4*)scale)[lane];
  float4 sf = ((const float4*)shift)[lane];
  acc.x = fmaxf(acc.x + bi.x, 0.f) * sc.x + sf.x;
  acc.y = fmaxf(acc.y + bi.y, 0.f) * sc.y + sf.y;
  acc.z = fmaxf(acc.z + bi.z, 0.f) * sc.z + sf.z;
  acc.w = fmaxf(acc.w + bi.w, 0.f) * sc.w + sf.w;
  ((float4*)(h + (size_t)v * HID))[lane] = acc;
}

// ======================= pooling + head =======================

// start[g] = lower_bound(batch, g), g in [0, G]; batch is sorted.
__global__ void k_gstart(const int* __restrict__ batch, int N, int G, int* __restrict__ start) {
  int g = blockIdx.x * blockDim.x + threadIdx.x;
  if (g > G) return;
  int lo = 0, hi = N;
  while (lo < hi) {
    int mid = (lo + hi) >> 1;
    if (batch[mid] < g) lo = mid + 1; else hi = mid;
  }
  start[g] = lo;
}

// Deterministic per-graph mean pool: one wave per graph.
__global__ __launch_bounds__(256) void k_pool(const float* __restrict__ h,
                                              const int* __restrict__ start,
                                              float* __restrict__ pooled, int G) {
  int wave = threadIdx.x >> 5, lane = threadIdx.x & 31;
  int g = blockIdx.x * 8 + wave;
  if (g >= G) return;
  int beg = start[g], end = start[g + 1];
  float4 acc = {0.f, 0.f, 0.f, 0.f};
  for (int i = beg; i < end; ++i) {
    float4 r = ((const float4*)(h + (size_t)i * HID))[lane];
    acc.x += r.x; acc.y += r.y; acc.z += r.z; acc.w += r.w;
  }
  float inv = 1.0f / fmaxf((float)(end - beg), 1.0f);
  acc.x *= inv; acc.y *= inv; acc.z *= inv; acc.w *= inv;
  ((float4*)(pooled + (size_t)g * HID))[lane] = acc;
}

// out[g] = relu(pooled[g] @ W1 + b1) @ W2 + b2   (one block per graph)
__global__ __launch_bounds__(128) void k_head(const float* __restrict__ pooled,
                                              const float* __restrict__ W1,
                                              const float* __restrict__ b1,
                                              const float* __restrict__ W2,
                                              const float* __restrict__ b2,
                                              float* __restrict__ out, int G) {
  __shared__ float ps[HID];
  __shared__ float red[HID];
  int g = blockIdx.x;
  int j = threadIdx.x;
  ps[j] = pooled[(size_t)g * HID + j];
  __syncthreads();
  float acc = b1[j];
#pragma unroll 8
  for (int k = 0; k < HID; ++k) acc = fmaf(ps[k], W1[(size_t)k * HID + j], acc);
  red[j] = fmaxf(acc, 0.f) * W2[j];
  __syncthreads();
  for (int s = 64; s > 0; s >>= 1) {
    if (j < s) red[j] += red[j + s];
    __syncthreads();
  }
  if (j == 0) out[g] = red[0] + b2[0];
}

// ======================= launcher =======================

extern "C" void kernel_launch(void* const* d_in, const int* in_sizes, int n_in,
                              void* d_out, int out_size, void* d_ws, size_t ws_size,
                              hipStream_t stream) {
  (void)n_in; (void)ws_size;
  const float* x   = (const float*)d_in[0];
  const int*   ei  = (const int*)d_in[1];   // [2, E], int32 per harness mapping
  const int*   bat = (const int*)d_in[2];   // [N], sorted
  const float* Ws  = (const float*)d_in[3]; // [4,128,128]
  const float* bs  = (const float*)d_in[4]; // [4,128]
  const float* gam = (const float*)d_in[5];
  const float* bet = (const float*)d_in[6];
  const float* bmn = (const float*)d_in[7];
  const float* bvr = (const float*)d_in[8];
  const float* hW1 = (const float*)d_in[9];
  const float* hb1 = (const float*)d_in[10];
  const float* hW2 = (const float*)d_in[11];
  const float* hb2 = (const float*)d_in[12];
  float* out = (float*)d_out;

  int N = in_sizes[0] / HID;  // 100000
  int E = in_sizes[1] / 2;    // 1600000
  int G = out_size;           // 4096

  // carve scratch (256B-aligned)
  char* p = (char*)d_ws;
  auto carve = [&](size_t bytes) -> char* {
    char* r = p; p += (bytes + 255) & ~(size_t)255; return r;
  };
  float* t       = (float*)carve((size_t)N * HID * 4);
  float* h       = (float*)carve((size_t)N * HID * 4);
  float* dinv    = (float*)carve((size_t)N * 4);
  float* selfn   = (float*)carve((size_t)N * 4);
  int*   counts  = (int*)carve((size_t)N * 4);
  int*   offsets = (int*)carve((size_t)(N + 1) * 4);
  int*   cursor  = (int*)carve((size_t)N * 4);
  int*   csr_src = (int*)carve((size_t)E * 4);
  float* csr_w   = (float*)carve((size_t)E * 4);
  int*   gstart  = (int*)carve((size_t)(G + 1) * 4);
  float* pooled  = (float*)carve((size_t)G * HID * 4);
  float* scale   = (float*)carve((size_t)4 * HID * 4);
  float* shift   = (float*)carve((size_t)4 * HID * 4);

  hipMemsetAsync(counts, 0, (size_t)N * 4, stream);
  hipMemsetAsync(cursor, 0, (size_t)N * 4, stream);

  k_hist<<<(E + 255) / 256, 256, 0, stream>>>(ei, E, counts);
  k_deg<<<(N + 255) / 256, 256, 0, stream>>>(counts, N, dinv, selfn);
  k_scan<<<1, 1024, 0, stream>>>(counts, offsets, N);
  k_fill<<<(E + 255) / 256, 256, 0, stream>>>(ei, E, offsets, cursor, dinv, csr_src, csr_w);
  k_bnprep<<<(4 * HID + 255) / 256, 256, 0, stream>>>(gam, bet, bmn, bvr, scale, shift);

  for (int L = 0; L < 4; ++L) {
    const float* hin = (L == 0) ? x : h;
    k_gemm<<<N / 16, 256, 0, stream>>>(hin, Ws + (size_t)L * HID * HID, t);
    k_aggregate<<<(N + 7) / 8, 256, 0, stream>>>(t, offsets, csr_src, csr_w, selfn,
                                                 bs + L * HID, scale + L * HID,
                                                 shift + L * HID, h, N);
  }

  k_gstart<<<(G + 1 + 255) / 256, 256, 0, stream>>>(bat, N, G, gstart);
  k_pool<<<(G + 7) / 8, 256, 0, stream>>>(h, gstart, pooled, G);
  k_head<<<G, 128, 0, stream>>>(pooled, hW1, hb1, hW2, hb2, out, G);
}